// CSWinAttentionBlock_76252849373542
// MI455X (gfx1250) — compile-verified
//
#include <hip/hip_runtime.h>
#include <math.h>

typedef __attribute__((ext_vector_type(2))) float v2f;
typedef __attribute__((ext_vector_type(8))) float v8f;

#define SSZ   7
#define NHEAD 8
#define CDIM  192
#define HDIM  24
#define HWDIM 112
#define NPIX  12544        // 112*112
#define LTOK  784          // 7*112 = 49*16
#define ASCALE 0.20412414523193154f  // 24^-0.5

// ---------------------------------------------------------------------------
// WMMA helper: D = A(16x4) * B(4x16) + C, exact fp32 on the matrix pipe.
// A layout : lane(half,idx): a = {A[idx][k0+2*half], A[idx][k0+2*half+1]}
// B layout : lane(half,idx): b = {B[k0+2*half][idx], B[k0+2*half+1][idx]}
// C/D      : VGPR r, lane(half,idx) -> D[r + 8*half][idx]
// ---------------------------------------------------------------------------
__device__ inline v8f wmma4(v2f a, v2f b, v8f c) {
  return __builtin_amdgcn_wmma_f32_16x16x4_f32(false, a, false, b, (short)0, c,
                                               false, false);
}

// ---------------------------------------------------------------------------
// out(M,N) = W(M,K) * X(K,N) + bias(M).  8 waves/block, 1 tile per wave.
// ---------------------------------------------------------------------------
__global__ __launch_bounds__(256) void gemm_bias_kernel(
    const float* __restrict__ W, const float* __restrict__ bias,
    const float* __restrict__ X, float* __restrict__ out,
    int M, int K, int N)
{
  const int wave = threadIdx.x >> 5;
  const int lane = threadIdx.x & 31;
  const int half = lane >> 4;
  const int idx  = lane & 15;
  const int nt = blockIdx.x * 8 + wave;
  const int mt = blockIdx.y;
  if (nt * 16 >= N) return;                 // wave-uniform, EXEC stays full

  const float* Wrow = W + (size_t)(mt * 16 + idx) * K;
  const float* Xcol = X + nt * 16 + idx;

  v8f acc = {};
  #pragma unroll 4
  for (int k = 0; k < K; k += 4) {
    const int kk = k + 2 * half;
    v2f a, b;
    a.x = Wrow[kk];
    a.y = Wrow[kk + 1];
    b.x = Xcol[(size_t)kk * N];
    b.y = Xcol[(size_t)(kk + 1) * N];
    acc = wmma4(a, b, acc);
  }
  #pragma unroll
  for (int r = 0; r < 8; ++r) {
    const int mrow = mt * 16 + r + 8 * half;
    out[(size_t)mrow * N + nt * 16 + idx] = acc[r] + bias[mrow];
  }
}

// token -> NCHW pixel index for a given stripe, per branch
__device__ inline int pix_of(int token, int stripe, int vertical) {
  if (vertical) {
    const int row = token / SSZ;
    const int c   = token - row * SSZ;
    return row * HWDIM + stripe * SSZ + c;
  }
  const int r = token / HWDIM;
  const int c = token - r * HWDIM;
  return (stripe * SSZ + r) * HWDIM + c;
}

// ---------------------------------------------------------------------------
// Flash-style attention.  grid = (49 row-tiles, 8 heads, 16 stripes),
// block = 1 wave (32 threads).  L = 784 = 49*16 exactly, no masking.
// qkv: (3*192, 12544) fp32;  out: (192, 12544) fp32 (NCHW pixel layout).
// ---------------------------------------------------------------------------
__global__ __launch_bounds__(32) void attn_kernel(
    const float* __restrict__ qkv, const float* __restrict__ table,
    float* __restrict__ out, int vertical)
{
  __shared__ float Pt[16 * 16];

  const int lane   = threadIdx.x & 31;
  const int half   = lane >> 4;
  const int idx    = lane & 15;
  const int mtile  = blockIdx.x;   // query row tile
  const int head   = blockIdx.y;
  const int stripe = blockIdx.z;

  const float* Qb = qkv + (size_t)(0 * CDIM + head * HDIM) * NPIX;
  const float* Kb = qkv + (size_t)(1 * CDIM + head * HDIM) * NPIX;
  const float* Vb = qkv + (size_t)(2 * CDIM + head * HDIM) * NPIX;

  // Preload this lane's slice of the Q tile: row = 16*mtile+idx,
  // k = 4*ks + 2*half + {0,1}, ks = 0..5 covers HD = 24 exactly.
  const int qpix = pix_of(16 * mtile + idx, stripe, vertical);
  float qa[12];
  #pragma unroll
  for (int ks = 0; ks < 6; ++ks) {
    const int kk = 4 * ks + 2 * half;
    qa[2 * ks]     = Qb[(size_t)kk * NPIX + qpix];
    qa[2 * ks + 1] = Qb[(size_t)(kk + 1) * NPIX + qpix];
  }

  float mstate[8], lstate[8];
  #pragma unroll
  for (int r = 0; r < 8; ++r) { mstate[r] = -1e30f; lstate[r] = 0.0f; }
  v8f o0 = {};   // output dims 0..15
  v8f o1 = {};   // output dims 16..23 (padded to 32)

  for (int jt = 0; jt < 49; ++jt) {
    // ---- S = Q * K^T for this 16x16 tile (6 WMMA K-steps over HD=24) ----
    const int kcol = 16 * jt + idx;
    const int kpix = pix_of(kcol, stripe, vertical);
    v8f s = {};
    #pragma unroll
    for (int ks = 0; ks < 6; ++ks) {
      const int kk = 4 * ks + 2 * half;
      v2f a, b;
      a.x = qa[2 * ks];
      a.y = qa[2 * ks + 1];
      b.x = Kb[(size_t)kk * NPIX + kpix];
      b.y = Kb[(size_t)(kk + 1) * NPIX + kpix];
      s = wmma4(a, b, s);
    }

    // ---- scale + relative bias + online softmax (row = r + 8*half) ----
    float alpha[8];
    #pragma unroll
    for (int r = 0; r < 8; ++r) {
      const int i = 16 * mtile + r + 8 * half;   // query token
      const int j = kcol;                        // key token (per lane)
      float bias;
      if (vertical)
        bias = table[(6 * 13 + ((i % SSZ) - (j % SSZ) + 6)) * NHEAD + head];
      else
        bias = table[(((i / HWDIM) - (j / HWDIM) + 6) * 13 + 6) * NHEAD + head];
      const float sv = s[r] * ASCALE + bias;
      float mx = sv;
      #pragma unroll
      for (int off = 1; off < 16; off <<= 1)
        mx = fmaxf(mx, __shfl_xor(mx, off, 32));
      const float newm = fmaxf(mstate[r], mx);
      const float p = __expf(sv - newm);
      float ps = p;
      #pragma unroll
      for (int off = 1; off < 16; off <<= 1)
        ps += __shfl_xor(ps, off, 32);
      alpha[r]  = __expf(mstate[r] - newm);
      lstate[r] = lstate[r] * alpha[r] + ps;
      mstate[r] = newm;
      Pt[(r + 8 * half) * 16 + idx] = p;   // P[i_local][j_local]
    }
    __syncthreads();

    // rescale accumulated output by alpha (per query row)
    #pragma unroll
    for (int r = 0; r < 8; ++r) { o0[r] *= alpha[r]; o1[r] *= alpha[r]; }

    // ---- O += P(16x16) * V_j(16x24) : 4 K-steps, 2 N-tiles ----
    #pragma unroll
    for (int ks = 0; ks < 4; ++ks) {
      const int kk  = 4 * ks + 2 * half;
      const int vp0 = pix_of(16 * jt + kk, stripe, vertical);
      const int vp1 = pix_of(16 * jt + kk + 1, stripe, vertical);
      v2f a;
      a.x = Pt[idx * 16 + kk];
      a.y = Pt[idx * 16 + kk + 1];
      v2f b0, b1;
      b0.x = Vb[(size_t)idx * NPIX + vp0];
      b0.y = Vb[(size_t)idx * NPIX + vp1];
      o0 = wmma4(a, b0, o0);
      const int d1 = 16 + idx;
      b1.x = (d1 < HDIM) ? Vb[(size_t)d1 * NPIX + vp0] : 0.0f;
      b1.y = (d1 < HDIM) ? Vb[(size_t)d1 * NPIX + vp1] : 0.0f;
      o1 = wmma4(a, b1, o1);
    }
    __syncthreads();   // Pt reused next iteration
  }

  // ---- finalize (divide by l) and scatter back to NCHW ----
  #pragma unroll
  for (int r = 0; r < 8; ++r) {
    const int i    = 16 * mtile + r + 8 * half;
    const int opix = pix_of(i, stripe, vertical);
    const float inv = 1.0f / lstate[r];
    out[(size_t)(head * HDIM + idx) * NPIX + opix] = o0[r] * inv;
    const int d1 = 16 + idx;
    if (d1 < HDIM)
      out[(size_t)(head * HDIM + d1) * NPIX + opix] = o1[r] * inv;
  }
}

// ---------------------------------------------------------------------------
// LePE (depthwise 3x3 + exact GELU, residual) for both branches + average.
// pre = 0.5 * (lepe_h(attn_h) + lepe_v(attn_v))
// ---------------------------------------------------------------------------
__device__ inline float gelu_exact(float x) {
  return 0.5f * x * (1.0f + erff(x * 0.7071067811865476f));
}

__global__ __launch_bounds__(256) void lepe_combine_kernel(
    const float* __restrict__ ah, const float* __restrict__ av,
    const float* __restrict__ wh, const float* __restrict__ bh,
    const float* __restrict__ wv, const float* __restrict__ bv,
    float* __restrict__ pre)
{
  const int t = blockIdx.x * blockDim.x + threadIdx.x;
  if (t >= CDIM * NPIX) return;
  const int c = t / NPIX;
  const int p = t - c * NPIX;
  const int y = p / HWDIM;
  const int x = p - y * HWDIM;

  float sh = bh[c], sv = bv[c];
  #pragma unroll
  for (int dy = -1; dy <= 1; ++dy) {
    #pragma unroll
    for (int dx = -1; dx <= 1; ++dx) {
      const int yy = y + dy, xx = x + dx;
      if (yy >= 0 && yy < HWDIM && xx >= 0 && xx < HWDIM) {
        const int w_i = c * 9 + (dy + 1) * 3 + (dx + 1);
        const size_t a_i = (size_t)c * NPIX + yy * HWDIM + xx;
        sh += wh[w_i] * ah[a_i];
        sv += wv[w_i] * av[a_i];
      }
    }
  }
  const float vh = ah[t] + gelu_exact(sh);
  const float vv = av[t] + gelu_exact(sv);
  pre[t] = 0.5f * (vh + vv);
}

// ---------------------------------------------------------------------------
extern "C" void kernel_launch(void* const* d_in, const int* in_sizes, int n_in,
                              void* d_out, int out_size, void* d_ws, size_t ws_size,
                              hipStream_t stream) {
  const float* x        = (const float*)d_in[0];   // (1,192,112,112)
  const float* qkv_h_w  = (const float*)d_in[1];   // (576,192)
  const float* qkv_h_b  = (const float*)d_in[2];   // (576)
  const float* qkv_v_w  = (const float*)d_in[3];
  const float* qkv_v_b  = (const float*)d_in[4];
  const float* proj_w   = (const float*)d_in[5];   // (192,192)
  const float* proj_b   = (const float*)d_in[6];
  const float* lepe_h_w = (const float*)d_in[7];   // (192,1,3,3)
  const float* lepe_h_b = (const float*)d_in[8];
  const float* lepe_v_w = (const float*)d_in[9];
  const float* lepe_v_b = (const float*)d_in[10];
  const float* table_h  = (const float*)d_in[11];  // (13,13,8)
  const float* table_v  = (const float*)d_in[12];

  float* ws = (float*)d_ws;
  float* f_qkv_h = ws;                              // 576*12544
  float* f_qkv_v = f_qkv_h + (size_t)576 * NPIX;
  float* f_attn_h = f_qkv_v + (size_t)576 * NPIX;   // 192*12544
  float* f_attn_v = f_attn_h + (size_t)CDIM * NPIX;
  float* f_pre    = f_attn_v + (size_t)CDIM * NPIX;
  float* outp     = (float*)d_out;

  // 1-2. QKV projections (GEMM 576x192x12544), both branches
  {
    dim3 grid(98, 36, 1);   // 784/8 N-tile groups, 36 M-tiles
    gemm_bias_kernel<<<grid, 256, 0, stream>>>(qkv_h_w, qkv_h_b, x, f_qkv_h,
                                               576, CDIM, NPIX);
    gemm_bias_kernel<<<grid, 256, 0, stream>>>(qkv_v_w, qkv_v_b, x, f_qkv_v,
                                               576, CDIM, NPIX);
  }

  // 3-4. Striped attention, both branches
  {
    dim3 grid(49, NHEAD, 16);   // row tiles x heads x stripes
    attn_kernel<<<grid, 32, 0, stream>>>(f_qkv_h, table_h, f_attn_h, 0);
    attn_kernel<<<grid, 32, 0, stream>>>(f_qkv_v, table_v, f_attn_v, 1);
  }

  // 5. LePE + combine
  {
    const int total = CDIM * NPIX;
    lepe_combine_kernel<<<(total + 255) / 256, 256, 0, stream>>>(
        f_attn_h, f_attn_v, lepe_h_w, lepe_h_b, lepe_v_w, lepe_v_b, f_pre);
  }

  // 6. Final projection (GEMM 192x192x12544) into d_out
  {
    dim3 grid(98, 12, 1);
    gemm_bias_kernel<<<grid, 256, 0, stream>>>(proj_w, proj_b, f_pre, outp,
                                               CDIM, CDIM, NPIX);
  }
}